// TTT_DownProj_Wrapper_40750649704748
// MI455X (gfx1250) — compile-verified
//
#include <hip/hip_runtime.h>
#include <hip/hip_bf16.h>

// ---------------------------------------------------------------------------
// TTT-LoRA down-projection for MI455X (gfx1250, wave32).
// - All GEMMs use V_WMMA_F32_16X16X4_F32 (problem is HBM-bound: ~0.5 GB
//   traffic vs ~1 TFLOP, so fp32 matrix ops keep reference precision free).
// - The dominant GEMM (x @ W_base^T) uses the Tensor Data Mover:
//   tensor_load_to_lds DMAs 64x32 tiles into LDS (descriptor-padded to a
//   36-float row stride for conflict-free 16-lane reads), double-buffered,
//   synchronized with s_wait_tensorcnt + workgroup barriers.
// ---------------------------------------------------------------------------

typedef float v2f __attribute__((ext_vector_type(2)));
typedef float v8f __attribute__((ext_vector_type(8)));
typedef unsigned int u32x4 __attribute__((ext_vector_type(4)));
typedef int i32x4 __attribute__((ext_vector_type(4)));
typedef int i32x8 __attribute__((ext_vector_type(8)));

#define BSZ   2
#define SEQ   2048
#define KD    11008
#define HD    4096
#define ED    4096
#define RK    16
#define CHK   64
#define NCPB  32            // chunks per batch
#define NCH   (BSZ * NCPB)  // 64 total chunks
#define BS    (BSZ * SEQ)   // 4096 flattened rows
#define LR    0.02f         // TTT_LR * SCALING
#define SCALE 2.0f

static __device__ __forceinline__ v8f wmma4(v2f a, v2f b, v8f c) {
    // D = A(16x4,f32) * B(4x16,f32) + C(16x16,f32)
    return __builtin_amdgcn_wmma_f32_16x16x4_f32(
        false, a, false, b, (short)0, c, false, false);
}

// ---------------------------------------------------------------------------
// TDM: 2D tile load global -> LDS.
//   tile = tile_d1 rows x tile_d0 elements (f32), row stride in memory =
//   stride0 elements.  LDS destination gets 4 DWORDs of padding after every
//   32 DWORDs (pad_interval=4, pad_amount=3) -> effective LDS row stride of
//   36 floats when tile_d0 == 32.
// ---------------------------------------------------------------------------
static __device__ __forceinline__ void tdm_load_tile_2d(
    unsigned lds_byte_off, const float* gptr,
    unsigned tensor_d0, unsigned tensor_d1,
    unsigned tile_d0, unsigned tile_d1, unsigned stride0) {
    unsigned long long ga = (unsigned long long)(const void*)gptr;
    u32x4 g0;
    g0[0] = 1u;                                     // count=1 (valid), user desc
    g0[1] = lds_byte_off;                           // LDS address (bytes)
    g0[2] = (unsigned)(ga & 0xFFFFFFFFu);           // global_addr[31:0]
    g0[3] = (unsigned)((ga >> 32) & 0x1FFFFFFu)     // global_addr[56:32]
            | (2u << 30);                           // type = 2 ("image")
    i32x8 g1;
    g1[0] = (2 << 16)        // data_size = 4 bytes
          | (1 << 20)        // pad_enable
          | (4 << 22)        // pad_interval: 32 DWORDs
          | (3 << 25);       // pad_amount : 4 DWORDs
    g1[1] = (int)((tensor_d0 & 0xFFFFu) << 16);                       // dim0 lo16
    g1[2] = (int)(((tensor_d0 >> 16) & 0xFFFFu) |
                  ((tensor_d1 & 0xFFFFu) << 16));                     // dim0 hi / dim1 lo
    g1[3] = (int)(((tensor_d1 >> 16) & 0xFFFFu) |
                  ((tile_d0 & 0xFFFFu) << 16));                       // dim1 hi / tile0
    g1[4] = (int)(tile_d1 & 0xFFFFu);                                 // tile1, tile2=0
    g1[5] = (int)stride0;                                             // stride0[31:0]
    g1[6] = 0;                                                        // stride0 hi, stride1 lo
    g1[7] = 0;
    i32x4 z4 = {0, 0, 0, 0};
    i32x8 z8 = {0, 0, 0, 0, 0, 0, 0, 0};
    // clang-23 / therock-10.0 arity: (g0, g1, g2, g3, g4, cpol)
    __builtin_amdgcn_tensor_load_to_lds(g0, g1, z4, z4, z8, 0);
}

// ---------------------------------------------------------------------------
// Tiled GEMM: out[M,N] = X[M,K] * W[N,K]^T.  64x64 block tile, K-tile 32,
// 256 threads = 8 waves, each wave owns two 16x16 WMMA accumulators.
// A/W tiles are moved by the Tensor Data Mover, double-buffered in LDS.
// ---------------------------------------------------------------------------
#define TM 64
#define TN 64
#define TK 32
#define LDT 36   // padded LDS row stride (TDM pad: 32 data + 4 pad DWORDs)

__global__ __launch_bounds__(256)
void gemm_base_kernel(const float* __restrict__ X, const float* __restrict__ W,
                      float* __restrict__ out, int K) {
    __shared__ float As[2][TM * LDT];
    __shared__ float Bs[2][TN * LDT];
    const int t    = threadIdx.x;
    const int wave = t >> 5;
    const int lane = t & 31;
    const int lm   = lane & 15;
    const int hi   = (lane >> 4) & 1;
    const int m0   = blockIdx.x * TM;
    const int n0   = blockIdx.y * TN;
    const int ni   = (wave & 3) * 16;
    const int mi0  = (wave >> 2) * 16;
    const int mi1  = mi0 + 32;
    const int nk   = K / TK;

    const unsigned aOff0 = (unsigned)(unsigned long long)(void*)&As[0][0];
    const unsigned aOff1 = (unsigned)(unsigned long long)(void*)&As[1][0];
    const unsigned bOff0 = (unsigned)(unsigned long long)(void*)&Bs[0][0];
    const unsigned bOff1 = (unsigned)(unsigned long long)(void*)&Bs[1][0];

    // Prologue: prime both LDS buffers (wave 0 drives the TDM).
    if (wave == 0) {
        tdm_load_tile_2d(aOff0, X + (size_t)m0 * K, K, TM, TK, TM, K);
        tdm_load_tile_2d(bOff0, W + (size_t)n0 * K, K, TN, TK, TN, K);
        if (nk > 1) {
            tdm_load_tile_2d(aOff1, X + (size_t)m0 * K + TK, K, TM, TK, TM, K);
            tdm_load_tile_2d(bOff1, W + (size_t)n0 * K + TK, K, TN, TK, TN, K);
        }
    }

    v8f acc0 = {}, acc1 = {};
    for (int i = 0; i < nk; ++i) {
        const int buf = i & 1;
        if (wave == 0) {
            // Tensor ops retire in order per wave: allowing 2 outstanding
            // means tile i (and everything before it) has landed.
            if (i + 1 < nk) __builtin_amdgcn_s_wait_tensorcnt(2);
            else            __builtin_amdgcn_s_wait_tensorcnt(0);
        }
        __syncthreads();                    // tile i visible to all waves

        const float* Asb = As[buf];
        const float* Bsb = Bs[buf];
        #pragma unroll
        for (int kk = 0; kk < TK; kk += 4) {
            int ka = kk + 2 * hi;
            v2f b, a0, a1;
            b.x  = Bsb[(ni  + lm) * LDT + ka];  b.y  = Bsb[(ni  + lm) * LDT + ka + 1];
            a0.x = Asb[(mi0 + lm) * LDT + ka];  a0.y = Asb[(mi0 + lm) * LDT + ka + 1];
            a1.x = Asb[(mi1 + lm) * LDT + ka];  a1.y = Asb[(mi1 + lm) * LDT + ka + 1];
            acc0 = wmma4(a0, b, acc0);
            acc1 = wmma4(a1, b, acc1);
        }
        __syncthreads();                    // tile i consumed -> buffer reusable
        if (wave == 0 && i + 2 < nk) {
            int k0 = (i + 2) * TK;
            tdm_load_tile_2d(buf ? aOff1 : aOff0, X + (size_t)m0 * K + k0,
                             K, TM, TK, TM, K);
            tdm_load_tile_2d(buf ? bOff1 : bOff0, W + (size_t)n0 * K + k0,
                             K, TN, TK, TN, K);
        }
    }
    #pragma unroll
    for (int v = 0; v < 8; ++v) {
        int col = n0 + ni + lm;
        out[(size_t)(m0 + mi0 + v + 8 * hi) * HD + col] = acc0[v];
        out[(size_t)(m0 + mi1 + v + 8 * hi) * HD + col] = acc1[v];
    }
}

// ---------------------------------------------------------------------------
// V GEMM: A rows are next-token embedding gathers (shift-by-1, zero pad) --
// gathers can't ride the TDM, so this keeps single-buffered manual staging.
// ---------------------------------------------------------------------------
__global__ __launch_bounds__(256)
void gemm_v_kernel(const float* __restrict__ Etab, const float* __restrict__ Wp,
                   const int* __restrict__ ids, float* __restrict__ V) {
    __shared__ float As[TM * LDT];
    __shared__ float Bs[TN * LDT];
    const int t    = threadIdx.x;
    const int wave = t >> 5;
    const int lane = t & 31;
    const int lm   = lane & 15;
    const int hi   = (lane >> 4) & 1;
    const int m0   = blockIdx.x * TM;
    const int n0   = blockIdx.y * TN;
    const int ni   = (wave & 3) * 16;
    const int mi0  = (wave >> 2) * 16;
    const int mi1  = mi0 + 32;

    v8f acc0 = {}, acc1 = {};
    for (int k0 = 0; k0 < ED; k0 += TK) {
        #pragma unroll
        for (int i = 0; i < 2; ++i) {
            int slot = t + (i << 8);
            int r    = slot >> 3;
            int c    = (slot & 7) << 2;
            int m    = m0 + r;
            int bb   = m >> 11;          // / SEQ
            int s    = m & (SEQ - 1);
            float4 va;
            if (s == SEQ - 1) {
                va = make_float4(0.f, 0.f, 0.f, 0.f);
            } else {
                int tok = ids[(bb << 11) + s + 1];
                va = *(const float4*)(Etab + (size_t)tok * ED + k0 + c);
            }
            float4 vb = *(const float4*)(Wp + (size_t)(n0 + r) * ED + k0 + c);
            *(float4*)(As + r * LDT + c) = va;
            *(float4*)(Bs + r * LDT + c) = vb;
        }
        __syncthreads();
        #pragma unroll
        for (int kk = 0; kk < TK; kk += 4) {
            int ka = kk + 2 * hi;
            v2f b, a0, a1;
            b.x  = Bs[(ni  + lm) * LDT + ka];  b.y  = Bs[(ni  + lm) * LDT + ka + 1];
            a0.x = As[(mi0 + lm) * LDT + ka];  a0.y = As[(mi0 + lm) * LDT + ka + 1];
            a1.x = As[(mi1 + lm) * LDT + ka];  a1.y = As[(mi1 + lm) * LDT + ka + 1];
            acc0 = wmma4(a0, b, acc0);
            acc1 = wmma4(a1, b, acc1);
        }
        __syncthreads();
    }
    #pragma unroll
    for (int v = 0; v < 8; ++v) {
        int col = n0 + ni + lm;
        V[(size_t)(m0 + mi0 + v + 8 * hi) * HD + col] = acc0[v];
        V[(size_t)(m0 + mi1 + v + 8 * hi) * HD + col] = acc1[v];
    }
}

// ---------------------------------------------------------------------------
// proj_in[m, r] = sum_k X[m,k] * B0[r,k]   (M=BS, N=16, K=KD)
// ---------------------------------------------------------------------------
__global__ __launch_bounds__(128)
void proj_in_kernel(const float* __restrict__ X, const float* __restrict__ B0,
                    float* __restrict__ P) {
    const int wave = threadIdx.x >> 5, lane = threadIdx.x & 31;
    const int lm = lane & 15, hi = (lane >> 4) & 1;
    const int row0 = (blockIdx.x * 4 + wave) * 16;
    v8f acc = {};
    for (int k = 0; k < KD; k += 4) {
        int ka = k + 2 * hi;
        v2f a, b;
        a.x = X[(size_t)(row0 + lm) * KD + ka];
        a.y = X[(size_t)(row0 + lm) * KD + ka + 1];
        b.x = B0[(size_t)lm * KD + ka];
        b.y = B0[(size_t)lm * KD + ka + 1];
        acc = wmma4(a, b, acc);
    }
    #pragma unroll
    for (int v = 0; v < 8; ++v)
        P[(size_t)(row0 + v + 8 * hi) * RK + lm] = acc[v];
}

// proj_err[m, r] = sum_h V[m,h] * A0[h,r]   (B operand stored [K,N])
__global__ __launch_bounds__(128)
void proj_err_kernel(const float* __restrict__ V, const float* __restrict__ A0,
                     float* __restrict__ P) {
    const int wave = threadIdx.x >> 5, lane = threadIdx.x & 31;
    const int lm = lane & 15, hi = (lane >> 4) & 1;
    const int row0 = (blockIdx.x * 4 + wave) * 16;
    v8f acc = {};
    for (int k = 0; k < HD; k += 4) {
        int ka = k + 2 * hi;
        v2f a, b;
        a.x = V[(size_t)(row0 + lm) * HD + ka];
        a.y = V[(size_t)(row0 + lm) * HD + ka + 1];
        b.x = A0[(size_t)ka * RK + lm];
        b.y = A0[(size_t)(ka + 1) * RK + lm];
        acc = wmma4(a, b, acc);
    }
    #pragma unroll
    for (int v = 0; v < 8; ++v)
        P[(size_t)(row0 + v + 8 * hi) * RK + lm] = acc[v];
}

// dA[chunk, h, r] = sum_c V[cb+c, h] * Pin[cb+c, r]   (K=64 within chunk)
__global__ __launch_bounds__(128)
void dA_kernel(const float* __restrict__ V, const float* __restrict__ Pin,
               float* __restrict__ dA) {
    const int wave = threadIdx.x >> 5, lane = threadIdx.x & 31;
    const int lm = lane & 15, hi = (lane >> 4) & 1;
    const int chunk = blockIdx.y;
    const int cb = chunk * CHK;
    const int m0 = blockIdx.x * 64 + wave * 16;    // h tile
    v8f acc = {};
    #pragma unroll 4
    for (int k = 0; k < CHK; k += 4) {
        int ka = cb + k + 2 * hi;
        v2f a, b;
        a.x = V[(size_t)ka * HD + m0 + lm];
        a.y = V[(size_t)(ka + 1) * HD + m0 + lm];
        b.x = Pin[(size_t)ka * RK + lm];
        b.y = Pin[(size_t)(ka + 1) * RK + lm];
        acc = wmma4(a, b, acc);
    }
    #pragma unroll
    for (int v = 0; v < 8; ++v)
        dA[((size_t)chunk * HD + m0 + v + 8 * hi) * RK + lm] = acc[v];
}

// dB[chunk, r, k] = sum_c Perr[cb+c, r] * X[cb+c, k]
__global__ __launch_bounds__(128)
void dB_kernel(const float* __restrict__ Perr, const float* __restrict__ X,
               float* __restrict__ dB) {
    const int wave = threadIdx.x >> 5, lane = threadIdx.x & 31;
    const int lm = lane & 15, hi = (lane >> 4) & 1;
    const int chunk = blockIdx.y;
    const int cb = chunk * CHK;
    const int n0 = blockIdx.x * 64 + wave * 16;    // k tile
    v8f acc = {};
    #pragma unroll 4
    for (int k = 0; k < CHK; k += 4) {
        int ka = cb + k + 2 * hi;
        v2f a, b;
        a.x = Perr[(size_t)ka * RK + lm];
        a.y = Perr[(size_t)(ka + 1) * RK + lm];
        b.x = X[(size_t)ka * KD + n0 + lm];
        b.y = X[(size_t)(ka + 1) * KD + n0 + lm];
        acc = wmma4(a, b, acc);
    }
    #pragma unroll
    for (int v = 0; v < 8; ++v)
        dB[((size_t)chunk * RK + v + 8 * hi) * KD + n0 + lm] = acc[v];
}

// In-place exclusive prefix-sum across chunks: buf layout [nb, nc, per].
__global__ void excl_cumsum_kernel(float* __restrict__ buf, long per, int nc, int nb) {
    long idx = (long)blockIdx.x * blockDim.x + threadIdx.x;
    if (idx >= (long)nb * per) return;
    long b = idx / per, e = idx % per;
    float* p = buf + (size_t)b * nc * per + e;
    float run = 0.f;
    for (int i = 0; i < nc; ++i) {
        float v = p[(size_t)i * per];
        p[(size_t)i * per] = run;
        run += v;
    }
}

// mid[m, r] = sum_k X[m,k] * (B0[r,k] - lr*accB[chunk,r,k]); one block per chunk.
__global__ __launch_bounds__(128)
void mid_kernel(const float* __restrict__ X, const float* __restrict__ B0,
                const float* __restrict__ accB, float* __restrict__ Mid) {
    const int wave = threadIdx.x >> 5, lane = threadIdx.x & 31;
    const int lm = lane & 15, hi = (lane >> 4) & 1;
    const int chunk = blockIdx.x;
    const int row0 = chunk * CHK + wave * 16;
    const float* aB = accB + (size_t)chunk * RK * KD;
    v8f acc = {};
    for (int k = 0; k < KD; k += 4) {
        int ka = k + 2 * hi;
        v2f a, b;
        a.x = X[(size_t)(row0 + lm) * KD + ka];
        a.y = X[(size_t)(row0 + lm) * KD + ka + 1];
        b.x = B0[(size_t)lm * KD + ka]     - LR * aB[(size_t)lm * KD + ka];
        b.y = B0[(size_t)lm * KD + ka + 1] - LR * aB[(size_t)lm * KD + ka + 1];
        acc = wmma4(a, b, acc);
    }
    #pragma unroll
    for (int v = 0; v < 8; ++v)
        Mid[(size_t)(row0 + v + 8 * hi) * RK + lm] = acc[v];
}

// out[m, h] += SCALE * sum_r mid[m,r] * (A0[h,r] - lr*accA[chunk,h,r])
__global__ __launch_bounds__(128)
void lora_kernel(const float* __restrict__ Mid, const float* __restrict__ A0,
                 const float* __restrict__ accA, float* __restrict__ out) {
    const int wave = threadIdx.x >> 5, lane = threadIdx.x & 31;
    const int lm = lane & 15, hi = (lane >> 4) & 1;
    const int m0 = blockIdx.x * 16;                 // rows in flattened B*S
    const int n0 = blockIdx.y * 64 + wave * 16;     // h tile
    const int chunk = m0 >> 6;
    const float* aA = accA + (size_t)chunk * HD * RK;
    v8f acc = {};
    #pragma unroll
    for (int k = 0; k < RK; k += 4) {
        int ka = k + 2 * hi;
        v2f a, b;
        a.x = Mid[(size_t)(m0 + lm) * RK + ka];
        a.y = Mid[(size_t)(m0 + lm) * RK + ka + 1];
        size_t o = (size_t)(n0 + lm) * RK;
        b.x = A0[o + ka]     - LR * aA[o + ka];
        b.y = A0[o + ka + 1] - LR * aA[o + ka + 1];
        acc = wmma4(a, b, acc);
    }
    #pragma unroll
    for (int v = 0; v < 8; ++v) {
        size_t idx = (size_t)(m0 + v + 8 * hi) * HD + n0 + lm;
        out[idx] += SCALE * acc[v];
    }
}

// ---------------------------------------------------------------------------
extern "C" void kernel_launch(void* const* d_in, const int* in_sizes, int n_in,
                              void* d_out, int out_size, void* d_ws, size_t ws_size,
                              hipStream_t stream) {
    const float* x      = (const float*)d_in[0];
    const float* W_base = (const float*)d_in[1];
    const float* embed  = (const float*)d_in[2];
    const float* W_proj = (const float*)d_in[3];
    const float* A0     = (const float*)d_in[4];
    const float* B0     = (const float*)d_in[5];
    const int*   ids    = (const int*)d_in[6];
    float* out = (float*)d_out;

    float* V    = (float*)d_ws;                     // [BS, HD]
    float* Pin  = V    + (size_t)BS * HD;           // [BS, RK]
    float* Perr = Pin  + (size_t)BS * RK;           // [BS, RK]
    float* Mid  = Perr + (size_t)BS * RK;           // [BS, RK]
    float* dAb  = Mid  + (size_t)BS * RK;           // [NCH, HD, RK]
    float* dBb  = dAb  + (size_t)NCH * HD * RK;     // [NCH, RK, KD]

    // 1. base_out = x @ W_base^T   -> d_out   (TDM double-buffered)
    gemm_base_kernel<<<dim3(BS / TM, HD / TN), 256, 0, stream>>>(x, W_base, out, KD);
    // 2. V = shifted_embed @ W_proj^T  -> ws
    gemm_v_kernel<<<dim3(BS / TM, HD / TN), 256, 0, stream>>>(embed, W_proj, ids, V);
    // 3. proj_in = Z @ B0^T ; proj_err = V @ A0
    proj_in_kernel<<<BS / 64, 128, 0, stream>>>(x, B0, Pin);
    proj_err_kernel<<<BS / 64, 128, 0, stream>>>(V, A0, Perr);
    // 4. per-chunk outer products
    dA_kernel<<<dim3(HD / 64, NCH), 128, 0, stream>>>(V, Pin, dAb);
    dB_kernel<<<dim3(KD / 64, NCH), 128, 0, stream>>>(Perr, x, dBb);
    // 5. exclusive prefix sums over chunks (in-place -> accA/accB)
    {
        long nA = (long)BSZ * HD * RK;
        excl_cumsum_kernel<<<(unsigned)((nA + 255) / 256), 256, 0, stream>>>(
            dAb, (long)HD * RK, NCPB, BSZ);
        long nB = (long)BSZ * RK * KD;
        excl_cumsum_kernel<<<(unsigned)((nB + 255) / 256), 256, 0, stream>>>(
            dBb, (long)RK * KD, NCPB, BSZ);
    }
    // 6. mid = Z @ B_eff^T ; out += SCALE * mid @ A_eff^T
    mid_kernel<<<NCH, 128, 0, stream>>>(x, B0, dBb, Mid);
    lora_kernel<<<dim3(BS / 16, HD / 64), 128, 0, stream>>>(Mid, A0, dAb, out);
}